// TargetAspectSupervisedContrastiveLoss_3341484556727
// MI455X (gfx1250) — compile-verified
//
#include <hip/hip_runtime.h>
#include <hip/hip_bf16.h>
#include <math.h>

// ---------------------------------------------------------------------------
// Supervised contrastive loss, B=8192, D=256, T=0.07
//   S = z z^T / T  (f32, via V_WMMA_F32_16X16X4_F32 -- exact f32 matmul)
//   pass1: sum_neg[i] = sum_j exp(S_ij) * [lab_i != lab_j]
//   pass2: row_loss[i] = (1/npos_i) * sum_{j pos} log1p(sum_neg_i * exp(-S_ij))
//   reduce: loss = mean over rows with npos>0
// ---------------------------------------------------------------------------

typedef float v2f __attribute__((ext_vector_type(2)));
typedef float v8f __attribute__((ext_vector_type(8)));

#define BSZ        8192
#define DIM        256
#define INV_T      14.285714285714286f   // 1/0.07
#define WAVES      4                     // waves per workgroup
#define TILES      2                     // 16-row tiles per wave
#define ROWS_WAVE  (TILES * 16)          // 32 rows per wave
#define ROWS_WG    (WAVES * ROWS_WAVE)   // 128 rows per workgroup
#define LDS_STRIDE 260                   // 256 + 4 pad: float4-aligned, conflict-free
#define SMEM_BYTES (WAVES * ROWS_WAVE * LDS_STRIDE * 4)

// f32 WMMA 16x16x4: per-lane A fragment (m = lane%16, k = kstep*4 + (lane/16)*2 + e)
// and B fragment (n = lane%16, same k offsets) have identical addressing into
// row-major z, since S = z z^T.  C/D layout: c[v] = S[v + 8*(lane/16)][lane%16].

__device__ __forceinline__ v8f wmma_f32_k4(v2f a, v2f b, v8f c) {
    return __builtin_amdgcn_wmma_f32_16x16x4_f32(
        /*neg_a=*/false, a, /*neg_b=*/false, b,
        /*c_mod=*/(short)0, c, /*reuse_a=*/false, /*reuse_b=*/false);
}

__device__ __forceinline__ void stage_A(const float* __restrict__ z,
                                        float* __restrict__ myA,
                                        int rb, int lane) {
    // copy this wave's 32 rows of z into padded LDS (float4 granules)
    const int f4_per_row = DIM / 4;   // 64
    for (int i = lane; i < ROWS_WAVE * f4_per_row; i += 32) {
        int row = i >> 6;              // i / 64
        int kk  = (i & 63) << 2;       // (i % 64) * 4
        float4 v = *(const float4*)(z + (size_t)(rb + row) * DIM + kk);
        *(float4*)(myA + row * LDS_STRIDE + kk) = v;
    }
}

__device__ __forceinline__ float half_reduce16(float x) {
    // sum across the 16 lanes of each wave32 half (xor masks stay in-half)
    x += __shfl_xor(x, 8, 32);
    x += __shfl_xor(x, 4, 32);
    x += __shfl_xor(x, 2, 32);
    x += __shfl_xor(x, 1, 32);
    return x;
}

// ---------------------------- pass 1: sum_neg ------------------------------
__global__ void scl_pass1_sumneg(const float* __restrict__ z,
                                 const int* __restrict__ labels,
                                 float* __restrict__ sum_neg) {
    extern __shared__ float ldsA[];
    const int lane = threadIdx.x & 31;
    const int wave = threadIdx.x >> 5;
    const int half = lane >> 4;
    const int ln   = lane & 15;

    const int rb = (blockIdx.x * WAVES + wave) * ROWS_WAVE;
    float* myA = ldsA + wave * ROWS_WAVE * LDS_STRIDE;
    stage_A(z, myA, rb, lane);
    __syncthreads();

    int   rlab[TILES][8];
    float sacc[TILES][8];
    #pragma unroll
    for (int t = 0; t < TILES; ++t)
        #pragma unroll
        for (int v = 0; v < 8; ++v) {
            rlab[t][v] = labels[rb + t * 16 + v + 8 * half];
            sacc[t][v] = 0.0f;
        }

    for (int ct = 0; ct < BSZ / 16; ++ct) {
        const int col  = ct * 16 + ln;
        const int clab = labels[col];
        const float* zc = z + (size_t)col * DIM;

        v8f c0 = {0.f,0.f,0.f,0.f,0.f,0.f,0.f,0.f};
        v8f c1 = {0.f,0.f,0.f,0.f,0.f,0.f,0.f,0.f};
        #pragma unroll 8
        for (int k = 0; k < DIM / 4; ++k) {
            const int kb = k * 4 + half * 2;
            v2f bf = *(const v2f*)(zc + kb);
            v2f a0 = *(const v2f*)(myA + (0 * 16 + ln) * LDS_STRIDE + kb);
            v2f a1 = *(const v2f*)(myA + (1 * 16 + ln) * LDS_STRIDE + kb);
            c0 = wmma_f32_k4(a0, bf, c0);
            c1 = wmma_f32_k4(a1, bf, c1);
        }
        #pragma unroll
        for (int v = 0; v < 8; ++v) {
            float e0 = __expf(c0[v] * INV_T);
            float e1 = __expf(c1[v] * INV_T);
            sacc[0][v] += (rlab[0][v] != clab) ? e0 : 0.0f;
            sacc[1][v] += (rlab[1][v] != clab) ? e1 : 0.0f;
        }
    }

    #pragma unroll
    for (int t = 0; t < TILES; ++t)
        #pragma unroll
        for (int v = 0; v < 8; ++v) {
            float s = half_reduce16(sacc[t][v]);
            if (ln == 0) sum_neg[rb + t * 16 + v + 8 * half] = s;
        }
}

// ------------------------ pass 2: per-row loss -----------------------------
__global__ void scl_pass2_rowloss(const float* __restrict__ z,
                                  const int* __restrict__ labels,
                                  const float* __restrict__ sum_neg,
                                  float* __restrict__ row_loss,
                                  float* __restrict__ valid) {
    extern __shared__ float ldsA[];
    const int lane = threadIdx.x & 31;
    const int wave = threadIdx.x >> 5;
    const int half = lane >> 4;
    const int ln   = lane & 15;

    const int rb = (blockIdx.x * WAVES + wave) * ROWS_WAVE;
    float* myA = ldsA + wave * ROWS_WAVE * LDS_STRIDE;
    stage_A(z, myA, rb, lane);
    __syncthreads();

    int   rlab[TILES][8], ridx[TILES][8];
    float snr[TILES][8], lacc[TILES][8], cacc[TILES][8];
    #pragma unroll
    for (int t = 0; t < TILES; ++t)
        #pragma unroll
        for (int v = 0; v < 8; ++v) {
            int r = rb + t * 16 + v + 8 * half;
            ridx[t][v] = r;
            rlab[t][v] = labels[r];
            snr [t][v] = sum_neg[r];
            lacc[t][v] = 0.0f;
            cacc[t][v] = 0.0f;
        }

    for (int ct = 0; ct < BSZ / 16; ++ct) {
        const int col  = ct * 16 + ln;
        const int clab = labels[col];
        const float* zc = z + (size_t)col * DIM;

        v8f c0 = {0.f,0.f,0.f,0.f,0.f,0.f,0.f,0.f};
        v8f c1 = {0.f,0.f,0.f,0.f,0.f,0.f,0.f,0.f};
        #pragma unroll 8
        for (int k = 0; k < DIM / 4; ++k) {
            const int kb = k * 4 + half * 2;
            v2f bf = *(const v2f*)(zc + kb);
            v2f a0 = *(const v2f*)(myA + (0 * 16 + ln) * LDS_STRIDE + kb);
            v2f a1 = *(const v2f*)(myA + (1 * 16 + ln) * LDS_STRIDE + kb);
            c0 = wmma_f32_k4(a0, bf, c0);
            c1 = wmma_f32_k4(a1, bf, c1);
        }
        #pragma unroll
        for (int v = 0; v < 8; ++v) {
            bool p0 = (rlab[0][v] == clab) && (ridx[0][v] != col);
            bool p1 = (rlab[1][v] == clab) && (ridx[1][v] != col);
            // log1p(0) == 0 -> masking via the argument is exact
            float a0 = p0 ? snr[0][v] * __expf(-c0[v] * INV_T) : 0.0f;
            float a1 = p1 ? snr[1][v] * __expf(-c1[v] * INV_T) : 0.0f;
            lacc[0][v] += log1pf(a0);
            lacc[1][v] += log1pf(a1);
            cacc[0][v] += p0 ? 1.0f : 0.0f;
            cacc[1][v] += p1 ? 1.0f : 0.0f;
        }
    }

    #pragma unroll
    for (int t = 0; t < TILES; ++t)
        #pragma unroll
        for (int v = 0; v < 8; ++v) {
            float l = half_reduce16(lacc[t][v]);
            float c = half_reduce16(cacc[t][v]);
            if (ln == 0) {
                int r = rb + t * 16 + v + 8 * half;
                row_loss[r] = (c > 0.0f) ? l / c : 0.0f;
                valid[r]    = (c > 0.0f) ? 1.0f : 0.0f;
            }
        }
}

// ----------------------- final deterministic reduce ------------------------
__global__ void scl_reduce(const float* __restrict__ row_loss,
                           const float* __restrict__ valid,
                           float* __restrict__ out) {
    __shared__ float sl[256];
    __shared__ float sv[256];
    const int t = threadIdx.x;
    float a = 0.0f, b = 0.0f;
    for (int i = t; i < BSZ; i += 256) { a += row_loss[i]; b += valid[i]; }
    sl[t] = a; sv[t] = b;
    __syncthreads();
    for (int off = 128; off > 0; off >>= 1) {
        if (t < off) { sl[t] += sl[t + off]; sv[t] += sv[t + off]; }
        __syncthreads();
    }
    if (t == 0) {
        float nv = sv[0];
        out[0] = (nv > 0.0f) ? sl[0] / nv : 0.0f;
    }
}

// ---------------------------------------------------------------------------
extern "C" void kernel_launch(void* const* d_in, const int* in_sizes, int n_in,
                              void* d_out, int out_size, void* d_ws, size_t ws_size,
                              hipStream_t stream) {
    (void)in_sizes; (void)n_in; (void)out_size; (void)ws_size;
    const float* z      = (const float*)d_in[0];
    const int*   labels = (const int*)d_in[1];
    float* out = (float*)d_out;

    float* sum_neg  = (float*)d_ws;            // [8192]
    float* row_loss = sum_neg + BSZ;           // [8192]
    float* valid    = row_loss + BSZ;          // [8192]

    dim3 grid(BSZ / ROWS_WG), block(WAVES * 32);
    scl_pass1_sumneg <<<grid, block, SMEM_BYTES, stream>>>(z, labels, sum_neg);
    scl_pass2_rowloss<<<grid, block, SMEM_BYTES, stream>>>(z, labels, sum_neg,
                                                           row_loss, valid);
    scl_reduce<<<1, 256, 0, stream>>>(row_loss, valid, out);
}